// CoxPHLossSorted_22840636080128
// MI455X (gfx1250) — compile-verified
//
#include <hip/hip_runtime.h>
#include <math.h>

#define NB   8192            // blocks in streaming passes
#define TPB  256             // threads per block (8 waves of 32)
#define PT   16              // elements per thread
#define TILE (TPB * PT)      // 4096 elements per block; NB*TILE == 33554432
#define EPSF 1e-7f

typedef __attribute__((ext_vector_type(2))) float v2f;
typedef __attribute__((ext_vector_type(8))) float v8f;
typedef __attribute__((ext_vector_type(4))) float vf4;
typedef __attribute__((ext_vector_type(4))) int   vi4;

// ---------------- Pass A: per-block max + sum(exp(x - blockmax)) ----------------
__global__ __launch_bounds__(TPB) void k_partial(const float* __restrict__ lh,
                                                 float* __restrict__ bmax,
                                                 float* __restrict__ bsum) {
    __shared__ float red[TPB];
    const int tid = threadIdx.x;
    const long long base = (long long)blockIdx.x * TILE + (long long)tid * PT;
    const vf4* p4 = (const vf4*)(lh + base);
    vf4 x0 = p4[0], x1 = p4[1], x2 = p4[2], x3 = p4[3];

    float xf[PT] = {x0.x, x0.y, x0.z, x0.w, x1.x, x1.y, x1.z, x1.w,
                    x2.x, x2.y, x2.z, x2.w, x3.x, x3.y, x3.z, x3.w};

    float m = xf[0];
#pragma unroll
    for (int j = 1; j < PT; ++j) m = fmaxf(m, xf[j]);
    red[tid] = m;
    __syncthreads();
    for (int off = TPB / 2; off > 0; off >>= 1) {
        if (tid < off) red[tid] = fmaxf(red[tid], red[tid + off]);
        __syncthreads();
    }
    const float bm = red[0];
    __syncthreads();

    float s = 0.f;
#pragma unroll
    for (int j = 0; j < PT; ++j) s += expf(xf[j] - bm);
    red[tid] = s;
    __syncthreads();
    for (int off = TPB / 2; off > 0; off >>= 1) {
        if (tid < off) red[tid] += red[tid + off];
        __syncthreads();
    }
    if (tid == 0) { bmax[blockIdx.x] = bm; bsum[blockIdx.x] = red[0]; }
}

// -------- gamma = max(bmax); exclusive double scan of exp(bmax-gamma)*bsum --------
__global__ __launch_bounds__(TPB) void k_scan(const float* __restrict__ bmax,
                                              const float* __restrict__ bsum,
                                              double* __restrict__ bpref,
                                              double* __restrict__ gslot) {
    __shared__ double sd[TPB];
    const int tid = threadIdx.x;
    const int CH = NB / TPB;      // 32 entries per thread, contiguous
    const int b0 = tid * CH;

    float m = -INFINITY;
    for (int i = 0; i < CH; ++i) m = fmaxf(m, bmax[b0 + i]);
    sd[tid] = (double)m;
    __syncthreads();
    for (int off = TPB / 2; off > 0; off >>= 1) {
        if (tid < off) sd[tid] = fmax(sd[tid], sd[tid + off]);
        __syncthreads();
    }
    const float gamma = (float)sd[0];
    __syncthreads();

    double tot = 0.0;
    for (int i = 0; i < CH; ++i)
        tot += exp((double)bmax[b0 + i] - (double)gamma) * (double)bsum[b0 + i];
    sd[tid] = tot;
    __syncthreads();
    // Hillis-Steele inclusive scan (deterministic)
    for (int off = 1; off < TPB; off <<= 1) {
        double v = (tid >= off) ? sd[tid - off] : 0.0;
        __syncthreads();
        sd[tid] += v;
        __syncthreads();
    }
    double run = (tid == 0) ? 0.0 : sd[tid - 1];
    for (int i = 0; i < CH; ++i) {
        bpref[b0 + i] = run;
        run += exp((double)bmax[b0 + i] - (double)gamma) * (double)bsum[b0 + i];
    }
    if (tid == 0) gslot[0] = (double)gamma;
}

// ---------------- Pass B: fused scan + loss accumulation (WMMA block scan) ----------------
__global__ __launch_bounds__(TPB) void k_main(const float* __restrict__ lh,
                                              const int* __restrict__ ev,
                                              const double* __restrict__ bpref,
                                              const double* __restrict__ gslot,
                                              double* __restrict__ bterm,
                                              double* __restrict__ bevt) {
    __shared__ float  tvals[TPB];   // per-thread exp-sum totals (the 16x16 matrix T)
    __shared__ float  pref[TPB];    // exclusive prefix of thread totals
    __shared__ float  rs[16];       // 16 row totals of T
    __shared__ double sred[TPB];

    const int tid = threadIdx.x;
    const long long base = (long long)blockIdx.x * TILE + (long long)tid * PT;

    const vf4* p4 = (const vf4*)(lh + base);
    vf4 x0 = p4[0], x1 = p4[1], x2 = p4[2], x3 = p4[3];
    const vi4* e4 = (const vi4*)(ev + base);
    vi4 e0 = __builtin_nontemporal_load(e4 + 0);   // keep log_h resident in L2
    vi4 e1 = __builtin_nontemporal_load(e4 + 1);
    vi4 e2 = __builtin_nontemporal_load(e4 + 2);
    vi4 e3 = __builtin_nontemporal_load(e4 + 3);

    float xf[PT] = {x0.x, x0.y, x0.z, x0.w, x1.x, x1.y, x1.z, x1.w,
                    x2.x, x2.y, x2.z, x2.w, x3.x, x3.y, x3.z, x3.w};
    int   ei[PT] = {e0.x, e0.y, e0.z, e0.w, e1.x, e1.y, e1.z, e1.w,
                    e2.x, e2.y, e2.z, e2.w, e3.x, e3.y, e3.z, e3.w};

    const float  gamma = (float)gslot[0];
    const double P     = bpref[blockIdx.x];

    float ex[PT];
    float ttot = 0.f;
#pragma unroll
    for (int j = 0; j < PT; ++j) { ex[j] = expf(xf[j] - gamma); ttot += ex[j]; }
    tvals[tid] = ttot;
    __syncthreads();

    // --- wave 0: exclusive scan of tvals[256] via V_WMMA_F32_16X16X4_F32 ---
    // T[r][c] = tvals[16r+c].  colpref = T * U_strict (chain of 4 K=4 WMMAs).
    if (__builtin_amdgcn_readfirstlane(tid >> 5) == 0) {
        const int l    = tid;            // lane 0..31, EXEC all ones here
        const int mrow = l & 15;         // A: M = lane%16
        const int koff = (l >> 4) << 1;  // A/B: K = vgpr + 2*(lane/16)
        v8f acc = {};
#pragma unroll
        for (int k = 0; k < 4; ++k) {
            v2f a, b;
            a.x = tvals[16 * mrow + 4 * k + koff + 0];
            a.y = tvals[16 * mrow + 4 * k + koff + 1];
            // B = U_strict[4k .. 4k+3, :], constant triangular ones
            b.x = ((4 * k + koff + 0) < mrow) ? 1.0f : 0.0f;
            b.y = ((4 * k + koff + 1) < mrow) ? 1.0f : 0.0f;
            acc = __builtin_amdgcn_wmma_f32_16x16x4_f32(
                false, a, false, b, (short)0, acc, false, false);
        }
        // acc[g] = exclusive row prefix at (r = g + 8*(l>>4), c = l&15).
        // Row totals from the c==15 lanes: rowtot[r] = colpref[r][15] + T[r][15]
        if ((l & 15) == 15) {
            const int rb = (l >> 4) * 8;
#pragma unroll
            for (int g = 0; g < 8; ++g) {
                const int r = rb + g;
                rs[r] = acc[g] + tvals[16 * r + 15];
            }
        }
        asm volatile("s_wait_dscnt 0" ::: "memory");  // intra-wave LDS RAW
        float rv[16];
#pragma unroll
        for (int r = 0; r < 16; ++r) rv[r] = rs[r];
        float rp[16];
        rp[0] = 0.f;
#pragma unroll
        for (int r = 1; r < 16; ++r) rp[r] = rp[r - 1] + rv[r - 1];
        const int hi = (l >> 4) * 8;
        const int c  = l & 15;
#pragma unroll
        for (int g = 0; g < 8; ++g) {
            const int r = g + hi;
            pref[16 * r + c] = acc[g] + rp[r];   // exclusive prefix over thread ids
        }
    }
    __syncthreads();

    const double cumbase = P + (double)pref[tid];
    double term = 0.0;
    int    evc  = 0;
    float  runi = 0.f;
#pragma unroll
    for (int j = 0; j < PT; ++j) {
        runi += ex[j];                               // inclusive within thread
        const float cum = (float)(cumbase + (double)runi);
        const float lc  = logf(cum + EPSF) + gamma;  // log-cumsum-exp, stabilized
        term += (double)((xf[j] - lc) * (float)ei[j]);
        evc  += ei[j];
    }

    sred[tid] = term;
    __syncthreads();
    for (int off = TPB / 2; off > 0; off >>= 1) {
        if (tid < off) sred[tid] += sred[tid + off];
        __syncthreads();
    }
    if (tid == 0) bterm[blockIdx.x] = sred[0];
    __syncthreads();
    sred[tid] = (double)evc;
    __syncthreads();
    for (int off = TPB / 2; off > 0; off >>= 1) {
        if (tid < off) sred[tid] += sred[tid + off];
        __syncthreads();
    }
    if (tid == 0) bevt[blockIdx.x] = sred[0];
}

// ---------------- Final: loss = -sum(term)/sum(ev) ----------------
__global__ __launch_bounds__(TPB) void k_final(const double* __restrict__ bterm,
                                               const double* __restrict__ bevt,
                                               float* __restrict__ out) {
    __shared__ double sa[TPB], sb[TPB];
    const int tid = threadIdx.x;
    const int CH = NB / TPB;
    double t = 0.0, e = 0.0;
    for (int i = 0; i < CH; ++i) { t += bterm[tid * CH + i]; e += bevt[tid * CH + i]; }
    sa[tid] = t; sb[tid] = e;
    __syncthreads();
    for (int off = TPB / 2; off > 0; off >>= 1) {
        if (tid < off) { sa[tid] += sa[tid + off]; sb[tid] += sb[tid + off]; }
        __syncthreads();
    }
    if (tid == 0) out[0] = (float)(-sa[0] / sb[0]);
}

extern "C" void kernel_launch(void* const* d_in, const int* in_sizes, int n_in,
                              void* d_out, int out_size, void* d_ws, size_t ws_size,
                              hipStream_t stream) {
    (void)in_sizes; (void)n_in; (void)out_size; (void)ws_size;
    const float* lh = (const float*)d_in[0];
    const int*   ev = (const int*)d_in[1];

    // workspace layout (~256 KB)
    float*  bmax  = (float*)d_ws;
    float*  bsum  = bmax + NB;
    double* bpref = (double*)(bsum + NB);
    double* bterm = bpref + NB;
    double* bevt  = bterm + NB;
    double* gslot = bevt + NB;

    k_partial<<<NB, TPB, 0, stream>>>(lh, bmax, bsum);
    k_scan<<<1, TPB, 0, stream>>>(bmax, bsum, bpref, gslot);
    k_main<<<NB, TPB, 0, stream>>>(lh, ev, bpref, gslot, bterm, bevt);
    k_final<<<1, TPB, 0, stream>>>(bterm, bevt, (float*)d_out);
}